// semantic_80333068304393
// MI455X (gfx1250) — compile-verified
//
#include <hip/hip_runtime.h>
#include <hip/hip_bf16.h>

// Problem constants (match reference setup_inputs)
#define BVAL 16      // batch
#define CVAL 80      // classes
#define DVAL 1024    // feature dim
#define SP   196     // spatial positions (14*14)
#define K1_THREADS 224  // 7 waves; 196/224 = 87.5% active lanes in tanh loop

typedef __attribute__((ext_vector_type(2))) float v2f;
typedef __attribute__((ext_vector_type(8))) float v8f;

#if __has_builtin(__builtin_amdgcn_tanhf)
#define FAST_TANH(x) __builtin_amdgcn_tanhf(x)   // single v_tanh_f32 (trans pipe)
#else
#define FAST_TANH(x) tanhf(x)
#endif

// ---------------------------------------------------------------------------
// Kernel 1: scores + softmax -> coef[b, p, c]
// One block per (class, batch). Threads 0..195 each own one spatial position.
// word[c,:] and fc_a_w staged in LDS (broadcast ds reads in the hot loop).
// img[b,d,:] is contiguous in p -> perfectly coalesced global loads.
// fc_a_b is uniform over p and cancels in the softmax, so it is skipped.
// Dominant cost: B*P*C*D = 257M v_tanh_f32 -> maximize active-lane ratio.
// ---------------------------------------------------------------------------
__global__ __launch_bounds__(K1_THREADS) void attn_coef_kernel(
    const float* __restrict__ img,    // [B, D, SP]
    const float* __restrict__ word,   // [C, D]
    const float* __restrict__ fa_w,   // [D]
    float* __restrict__ coef)         // [B, SP, C]
{
  __shared__ float wc[DVAL];
  __shared__ float wa[DVAL];
  __shared__ float red[256];          // padded to power of two for the tree

  const int tid = threadIdx.x;        // 0..223
  const int c = blockIdx.x;
  const int b = blockIdx.y;

  for (int i = tid; i < DVAL; i += K1_THREADS) {
    wc[i] = word[(size_t)c * DVAL + i];
    wa[i] = fa_w[i];
  }
  __syncthreads();

  float acc = 0.0f;
  if (tid < SP) {
    const float* ip = img + ((size_t)b * DVAL) * SP + tid;
#pragma unroll 8
    for (int d = 0; d < DVAL; ++d) {
      acc += FAST_TANH(ip[(size_t)d * SP] * wc[d]) * wa[d];
    }
  }

  // ---- softmax over the 196 positions (block-wide tree reduction) ----
  red[tid] = (tid < SP) ? acc : -__builtin_inff();
  if (tid < 256 - K1_THREADS) red[K1_THREADS + tid] = -__builtin_inff();
  __syncthreads();
  for (int s = 128; s > 0; s >>= 1) {
    if (tid < s) red[tid] = fmaxf(red[tid], red[tid + s]);
    __syncthreads();
  }
  const float mx = red[0];
  __syncthreads();

  const float e = (tid < SP) ? __expf(acc - mx) : 0.0f;
  red[tid] = e;
  if (tid < 256 - K1_THREADS) red[K1_THREADS + tid] = 0.0f;
  __syncthreads();
  for (int s = 128; s > 0; s >>= 1) {
    if (tid < s) red[tid] += red[tid + s];
    __syncthreads();
  }
  const float inv = 1.0f / red[0];

  if (tid < SP) coef[((size_t)b * SP + tid) * CVAL + c] = e * inv;
}

// ---------------------------------------------------------------------------
// Kernel 2: out[b] (80x1024) = coef[b]^T (80x196) @ img[b] (196x1024)
// via V_WMMA_F32_16X16X4_F32, one wave per 16x16 tile, 49 k-steps.
//   A[m,k] = coef[b,k,m]   (m contiguous -> coalesced lane loads)
//   B[k,n] = img[b,n,k]    (k contiguous -> 8B per-lane loads)
// Fragment layouts per CDNA5 ISA 7.12.2:
//   A 16x4 : lanes 0-15 M=l K={k0,k0+1}; lanes 16-31 M=l K={k0+2,k0+3}
//   B 4x16 : lanes 0-15 N=l K={k0,k0+1}; lanes 16-31 N=l K={k0+2,k0+3}
//   C/D    : VGPR r -> (M=r, N=l) lanes 0-15, (M=r+8, N=l) lanes 16-31
// ---------------------------------------------------------------------------
__global__ __launch_bounds__(256) void attn_pool_wmma_kernel(
    const float* __restrict__ coef,   // [B, SP, C]
    const float* __restrict__ img,    // [B, D, SP]
    float* __restrict__ out)          // [B, C, D]
{
  const int lane = threadIdx.x & 31;
  const int wave = threadIdx.x >> 5;
  const bool hi = lane >= 16;
  const int l = lane & 15;

  const int b  = blockIdx.z;
  const int m0 = blockIdx.y * 16;                  // class tile
  const int n0 = (blockIdx.x * 8 + wave) * 16;     // feature tile (8 waves/block)

  const float* Abase = coef + (size_t)b * SP * CVAL + (m0 + l);
  const float* Bbase = img + ((size_t)b * DVAL + (n0 + l)) * SP;

  v8f acc = {};
  for (int kk = 0; kk < SP / 4; ++kk) {
    const int k0 = kk * 4 + (hi ? 2 : 0);

    v2f av;
    av.x = Abase[(size_t)(k0 + 0) * CVAL];
    av.y = Abase[(size_t)(k0 + 1) * CVAL];

    const float2 bt = *(const float2*)(Bbase + k0);  // 8B-aligned, k-contiguous
    v2f bv;
    bv.x = bt.x;
    bv.y = bt.y;

    acc = __builtin_amdgcn_wmma_f32_16x16x4_f32(
        /*neg_a=*/false, av, /*neg_b=*/false, bv,
        /*c_mod=*/(short)0, acc, /*reuse_a=*/false, /*reuse_b=*/false);
  }

#pragma unroll
  for (int r = 0; r < 8; ++r) {
    const int M = m0 + r + (hi ? 8 : 0);
    const int N = n0 + l;
    out[((size_t)b * CVAL + M) * DVAL + N] = acc[r];
  }
}

// ---------------------------------------------------------------------------
extern "C" void kernel_launch(void* const* d_in, const int* in_sizes, int n_in,
                              void* d_out, int out_size, void* d_ws, size_t ws_size,
                              hipStream_t stream) {
  // inputs: [0]=batch_size(int), [1]=img_feature_map f32 [B,D,S,S],
  //         [2]=word_features f32 [C,D], [3]=fc_a_w f32 [D], [4]=fc_a_b f32 (unused: cancels)
  const float* img  = (const float*)d_in[1];
  const float* word = (const float*)d_in[2];
  const float* faw  = (const float*)d_in[3];
  float* out  = (float*)d_out;
  float* coef = (float*)d_ws;   // B*SP*C floats = ~1.0 MB scratch

  dim3 g1(CVAL, BVAL);
  attn_coef_kernel<<<g1, K1_THREADS, 0, stream>>>(img, word, faw, coef);

  dim3 g2(DVAL / (16 * 8), CVAL / 16, BVAL);  // (8, 5, 16); 8 waves/block, 1 tile/wave
  attn_pool_wmma_kernel<<<g2, 256, 0, stream>>>(coef, img, out);
}